// BettiMatchingLoss_979252543807
// MI455X (gfx1250) — compile-verified
//
#include <hip/hip_runtime.h>

// ---------------------------------------------------------------------------
// BettiMatching MSE surrogate for MI455X (gfx1250, wave32).
//
// probs: (2,2,32,512,512) f32, gt: (2,32,512,512) i32 in {0,1,2}.
// Output scalar = global mean of (probs - onehot(gt))^2.
// Memory-bound: 192 MiB @ 23.3 TB/s ~ 8.6 us. Streaming b128 loads,
// f32 WMMA (16x16x4) used as an exact in-register reduction tree.
// ---------------------------------------------------------------------------

typedef float v2f __attribute__((ext_vector_type(2)));
typedef float v8f __attribute__((ext_vector_type(8)));

#define DHW_    (1u << 23)            // 32*512*512 spatial elements per (b,c)
#define PER_B_  (1u << 24)            // 2*DHW: probs elements per batch
#define NVEC_   (1u << 22)            // total voxels / 4 (float4 groups)
#define INV_N_  (1.0f / 33554432.0f)  // 1 / (2*2*32*512*512) == 2^-25

__device__ __forceinline__ float wave_reduce32(float x) {
#pragma unroll
  for (int off = 16; off > 0; off >>= 1) x += __shfl_xor(x, off, 32);
  return x;
}

__global__ void __launch_bounds__(256)
betti_mse_partial(const float* __restrict__ probs,
                  const int*   __restrict__ gt,
                  float*       __restrict__ partials) {
  const unsigned tid    = blockIdx.x * blockDim.x + threadIdx.x;
  const unsigned stride = gridDim.x * blockDim.x;
  const unsigned iters  = NVEC_ / stride;  // uniform trip count (EXEC all-1s)

  const v2f ones = {1.0f, 1.0f};
  v8f acc0 = {}, acc1 = {}, acc2 = {}, acc3 = {};
  float tail = 0.0f;

  const float4* __restrict__ p4 = (const float4*)probs;
  const int4*   __restrict__ g4 = (const int4*)gt;

  for (unsigned k = 0; k < iters; ++k) {
    const unsigned g    = tid + k * stride;       // float4-group index
    const unsigned v    = g << 2;                 // voxel index (b,d,h,w flat)
    const unsigned b    = v >> 23;                // batch
    const unsigned s    = v & (DHW_ - 1u);        // spatial offset within batch
    const unsigned base = (b * PER_B_ + s) >> 2;  // float4 index, channel 0

    const float4 p0 = p4[base];                   // channel 0
    const float4 p1 = p4[base + (DHW_ >> 2)];     // channel 1
    const int4   gm = g4[g];

    const float t1x = (gm.x == 1) ? 1.0f : 0.0f;
    const float t1y = (gm.y == 1) ? 1.0f : 0.0f;
    const float t1z = (gm.z == 1) ? 1.0f : 0.0f;
    const float t1w = (gm.w == 1) ? 1.0f : 0.0f;

    const float e0x = p0.x - (1.0f - t1x), e1x = p1.x - t1x;
    const float e0y = p0.y - (1.0f - t1y), e1y = p1.y - t1y;
    const float e0z = p0.z - (1.0f - t1z), e1z = p1.z - t1z;
    const float e0w = p0.w - (1.0f - t1w), e1w = p1.w - t1w;

    const v2f ax = {e0x * e0x, e1x * e1x};
    const v2f ay = {e0y * e0y, e1y * e1y};
    const v2f az = {e0z * e0z, e1z * e1z};
    const v2f aw = {e0w * e0w, e1w * e1w};

    // D[m][n] += sum_k A[m][k] * 1  -> each WMMA folds 64 squared errors
    // (x16 column replication, divided out below). 4 independent acc chains.
    acc0 = __builtin_amdgcn_wmma_f32_16x16x4_f32(false, ax, false, ones,
                                                 (short)0, acc0, false, false);
    acc1 = __builtin_amdgcn_wmma_f32_16x16x4_f32(false, ay, false, ones,
                                                 (short)0, acc1, false, false);
    acc2 = __builtin_amdgcn_wmma_f32_16x16x4_f32(false, az, false, ones,
                                                 (short)0, acc2, false, false);
    acc3 = __builtin_amdgcn_wmma_f32_16x16x4_f32(false, aw, false, ones,
                                                 (short)0, acc3, false, false);
  }

  // Guarded scalar tail (empty for the fixed shapes; keeps WMMA out of any
  // potentially divergent region).
  for (unsigned g = tid + iters * stride; g < NVEC_; g += stride) {
    const unsigned v    = g << 2;
    const unsigned b    = v >> 23;
    const unsigned s    = v & (DHW_ - 1u);
    const unsigned base = (b * PER_B_ + s) >> 2;
    const float4 p0 = p4[base];
    const float4 p1 = p4[base + (DHW_ >> 2)];
    const int4   gm = g4[g];
    const float t1x = (gm.x == 1) ? 1.0f : 0.0f;
    const float t1y = (gm.y == 1) ? 1.0f : 0.0f;
    const float t1z = (gm.z == 1) ? 1.0f : 0.0f;
    const float t1w = (gm.w == 1) ? 1.0f : 0.0f;
    float e;
    e = p0.x - (1.0f - t1x); tail += e * e;  e = p1.x - t1x; tail += e * e;
    e = p0.y - (1.0f - t1y); tail += e * e;  e = p1.y - t1y; tail += e * e;
    e = p0.z - (1.0f - t1z); tail += e * e;  e = p1.z - t1z; tail += e * e;
    e = p0.w - (1.0f - t1w); tail += e * e;  e = p1.w - t1w; tail += e * e;
  }

  // Sum of all 256 accumulator entries = 16 * (sum of squared errors fed in).
  float accsum = 0.0f;
#pragma unroll
  for (int i = 0; i < 8; ++i) accsum += acc0[i] + acc1[i] + acc2[i] + acc3[i];
  float lane = tail + accsum * (1.0f / 16.0f);

  lane = wave_reduce32(lane);

  __shared__ float smem[8];
  const unsigned wid = threadIdx.x >> 5, lid = threadIdx.x & 31u;
  if (lid == 0) smem[wid] = lane;
  __syncthreads();
  if (wid == 0) {
    float v2 = (lid < (blockDim.x >> 5)) ? smem[lid] : 0.0f;
    v2 = wave_reduce32(v2);
    if (lid == 0) partials[blockIdx.x] = v2;
  }
}

__global__ void __launch_bounds__(256)
betti_mse_finalize(const float* __restrict__ partials,
                   float*       __restrict__ out, int n) {
  float s = 0.0f;
  for (int i = threadIdx.x; i < n; i += blockDim.x) s += partials[i];
  s = wave_reduce32(s);

  __shared__ float smem[8];
  const unsigned wid = threadIdx.x >> 5, lid = threadIdx.x & 31u;
  if (lid == 0) smem[wid] = s;
  __syncthreads();
  if (wid == 0) {
    float v2 = (lid < (blockDim.x >> 5)) ? smem[lid] : 0.0f;
    v2 = wave_reduce32(v2);
    if (lid == 0) out[0] = v2 * INV_N_;
  }
}

extern "C" void kernel_launch(void* const* d_in, const int* in_sizes, int n_in,
                              void* d_out, int out_size, void* d_ws, size_t ws_size,
                              hipStream_t stream) {
  (void)in_sizes; (void)n_in; (void)out_size; (void)ws_size;
  const float* probs = (const float*)d_in[0];   // (2,2,32,512,512) f32
  const int*   gt    = (const int*)d_in[1];     // (2,32,512,512)  i32
  float*       ws    = (float*)d_ws;            // 1024 block partials
  float*       out   = (float*)d_out;           // scalar f32

  const int blocks = 1024, threads = 256;       // 8192 wave32 -> saturate HBM
  betti_mse_partial<<<blocks, threads, 0, stream>>>(probs, gt, ws);
  betti_mse_finalize<<<1, threads, 0, stream>>>(ws, out, blocks);
}